// FlexibleGNN_36627481101106
// MI455X (gfx1250) — compile-verified
//
#include <hip/hip_runtime.h>
#include <hip/hip_bf16.h>

// Problem constants (from reference)
#define NN   10000
#define EE   160000
#define PP   4000
#define NDIM 8
#define EDIM 3
#define HH   4
#define CC   64
#define DD1  256     // H*C

#define HBITS 19
#define HSIZE (1u << HBITS)
#define HEMPTY 0xFFFFFFFFu
#define ENC_NEG_INF 0x007FFFFFu  // encodeF(-inf)

typedef __attribute__((ext_vector_type(2))) float v2f;
typedef __attribute__((ext_vector_type(8))) float v8f;

// ---------- helpers ----------
__device__ __forceinline__ unsigned encodeF(float f) {
  unsigned u = __float_as_uint(f);
  return (u & 0x80000000u) ? ~u : (u | 0x80000000u);
}
__device__ __forceinline__ float decodeF(unsigned u) {
  return (u & 0x80000000u) ? __uint_as_float(u & 0x7FFFFFFFu) : __uint_as_float(~u);
}
__device__ __forceinline__ float eluf(float x) { return x > 0.f ? x : expm1f(x); }
__device__ __forceinline__ unsigned hashK(unsigned k) {
  k ^= k >> 16; k *= 0x7feb352du; k ^= k >> 15; k *= 0x846ca68bu; k ^= k >> 16; return k;
}

// ---------- generic f32 WMMA GEMM: C[M,Nw] = A[M,K] @ W[K,Nw] + bias (+ELU) ----------
// One wave computes one 16x16 output tile via V_WMMA_F32_16X16X4_F32.
// A-frag: lanes 0-15 hold row m0+lane with K = {k0,k0+1}; lanes 16-31 hold K = {k0+2,k0+3}.
// B-frag: mirrored (VGPR v holds K-row k0+v for lanes 0-15, k0+v+2 for lanes 16-31).
// C/D: VGPR v -> row m0 + v + 8*(lane>=16), col n0 + (lane&15).
__global__ void wmma_gemm_f32(const float* __restrict__ A, const float* __restrict__ W,
                              const float* __restrict__ bias, float* __restrict__ Cout,
                              int M, int K, int Nw, int act) {
  const int lane = threadIdx.x & 31;
  const int wave = threadIdx.x >> 5;
  const int wavesPerBlock = blockDim.x >> 5;
  const int tilesN = Nw >> 4;
  const int tilesM = M >> 4;
  const int tileId = blockIdx.x * wavesPerBlock + wave;
  if (tileId >= tilesM * tilesN) return;
  const int tm = tileId / tilesN, tn = tileId % tilesN;
  const int m0 = tm << 4, n0 = tn << 4;
  const int half = lane >> 4;       // 0: lanes 0-15, 1: lanes 16-31
  const int l16 = lane & 15;
  const int arow = m0 + l16;
  const int bcol = n0 + l16;

  v8f acc = {};
  for (int k0 = 0; k0 < K; k0 += 4) {
    v2f a, b;
    a.x = A[(size_t)arow * K + (k0 + 2 * half + 0)];
    a.y = A[(size_t)arow * K + (k0 + 2 * half + 1)];
    b.x = W[(size_t)(k0 + 2 * half + 0) * Nw + bcol];
    b.y = W[(size_t)(k0 + 2 * half + 1) * Nw + bcol];
    acc = __builtin_amdgcn_wmma_f32_16x16x4_f32(false, a, false, b, (short)0, acc,
                                                false, false);
  }
  const float bv = bias ? bias[bcol] : 0.0f;
#pragma unroll
  for (int v = 0; v < 8; ++v) {
    int row = m0 + v + 8 * half;
    float r = acc[v] + bv;
    if (act == 1) r = eluf(r);
    Cout[(size_t)row * Nw + bcol] = r;
  }
}

// ---------- init kernels ----------
__global__ void fill_u32(unsigned* p, unsigned v, int n) {
  int i = blockIdx.x * blockDim.x + threadIdx.x;
  if (i < n) p[i] = v;
}
__global__ void fill_f32(float* p, float v, int n) {
  int i = blockIdx.x * blockDim.x + threadIdx.x;
  if (i < n) p[i] = v;
}
__global__ void bias_broadcast(float* __restrict__ H, const float* __restrict__ b) {
  H[(size_t)blockIdx.x * DD1 + threadIdx.x] = b[threadIdx.x];
}
__global__ void elu_inplace(float* p, int n) {
  int i = blockIdx.x * blockDim.x + threadIdx.x;
  if (i < n) p[i] = eluf(p[i]);
}

// ---------- GATv2 edge pipeline ----------
// One warp per edge: m = lrelu(xl[src]+xr[dst]+eattr@We); logits[h] = <m_h, att_h>;
// segment-max via monotonic-encoded atomicMax.
__global__ void edge_logits(const float* __restrict__ xl, const float* __restrict__ xr,
                            const float* __restrict__ eattr, const float* __restrict__ We,
                            const float* __restrict__ att, const int* __restrict__ src,
                            const int* __restrict__ dst, float* __restrict__ logits,
                            unsigned* __restrict__ amaxU) {
  int e = blockIdx.x * (blockDim.x >> 5) + (threadIdx.x >> 5);
  int lane = threadIdx.x & 31;
  if (e >= EE) return;
  int s = src[e], d = dst[e];
  float ea0 = eattr[(size_t)e * 3 + 0];
  float ea1 = eattr[(size_t)e * 3 + 1];
  float ea2 = eattr[(size_t)e * 3 + 2];
  const float* pl = xl + (size_t)s * DD1;
  const float* pr = xr + (size_t)d * DD1;
#pragma unroll
  for (int h = 0; h < HH; ++h) {
    float part = 0.f;
#pragma unroll
    for (int cc = 0; cc < 2; ++cc) {
      int k = h * CC + lane + cc * 32;
      float ee = ea0 * We[k] + ea1 * We[DD1 + k] + ea2 * We[2 * DD1 + k];
      float m = pl[k] + pr[k] + ee;
      m = m > 0.f ? m : 0.2f * m;
      part += m * att[k];
    }
    for (int off = 16; off; off >>= 1) part += __shfl_down(part, off, 32);
    if (lane == 0) {
      logits[(size_t)e * HH + h] = part;
      atomicMax(&amaxU[d * HH + h], encodeF(part));
    }
  }
}

__global__ void edge_softmax_num(const int* __restrict__ dst, float* __restrict__ logits,
                                 const unsigned* __restrict__ amaxU, float* __restrict__ den) {
  int t = blockIdx.x * blockDim.x + threadIdx.x;
  if (t >= EE * HH) return;
  int e = t >> 2, h = t & 3;
  int d = dst[e];
  float ex = expf(logits[t] - decodeF(amaxU[d * HH + h]));
  logits[t] = ex;  // reuse buffer for numerator
  atomicAdd(&den[d * HH + h], ex);
}

// One 256-thread block per edge: H[dst] += alpha * xl[src]
__global__ void edge_aggregate(const int* __restrict__ src, const int* __restrict__ dst,
                               const float* __restrict__ xl, const float* __restrict__ ex,
                               const float* __restrict__ den, float* __restrict__ H) {
  int e = blockIdx.x;
  int k = threadIdx.x;             // 0..255
  int h = k >> 6;
  int s = src[e], d = dst[e];
  float alpha = ex[(size_t)e * HH + h] / (den[d * HH + h] + 1e-16f);
  atomicAdd(&H[(size_t)d * DD1 + k], alpha * xl[(size_t)s * DD1 + k]);
}

// ---------- physical-edge lookup via hash of (src*N+dst) -> min edge index ----------
__global__ void hash_insert(const int* __restrict__ src, const int* __restrict__ dst,
                            unsigned* __restrict__ keys, unsigned* __restrict__ vals) {
  int e = blockIdx.x * blockDim.x + threadIdx.x;
  if (e >= EE) return;
  unsigned key = (unsigned)src[e] * (unsigned)NN + (unsigned)dst[e];
  unsigned h = hashK(key) & (HSIZE - 1);
  while (true) {
    unsigned old = atomicCAS(&keys[h], HEMPTY, key);
    if (old == HEMPTY || old == key) { atomicMin(&vals[h], (unsigned)e); break; }
    h = (h + 1) & (HSIZE - 1);
  }
}

__global__ void phys_lookup(const int* __restrict__ pe, const unsigned* __restrict__ keys,
                            const unsigned* __restrict__ vals, const float* __restrict__ eattr,
                            float* __restrict__ efeat) {
  int p = blockIdx.x * blockDim.x + threadIdx.x;
  if (p >= PP) return;
  int i = pe[p * 2 + 0], j = pe[p * 2 + 1];
  unsigned q0 = (unsigned)i * (unsigned)NN + (unsigned)j;
  unsigned q1 = (unsigned)j * (unsigned)NN + (unsigned)i;
  unsigned best = 0xFFFFFFFFu;
#pragma unroll
  for (int t = 0; t < 2; ++t) {
    unsigned q = t ? q1 : q0;
    unsigned h = hashK(q) & (HSIZE - 1);
    while (true) {
      unsigned k = keys[h];
      if (k == q) { unsigned v = vals[h]; if (v < best) best = v; break; }
      if (k == HEMPTY) break;
      h = (h + 1) & (HSIZE - 1);
    }
  }
  if (best != 0xFFFFFFFFu) {
    efeat[p * 3 + 0] = eattr[(size_t)best * 3 + 0];
    efeat[p * 3 + 1] = eattr[(size_t)best * 3 + 1];
    efeat[p * 3 + 2] = eattr[(size_t)best * 3 + 2];
  } else {
    efeat[p * 3 + 0] = 0.f; efeat[p * 3 + 1] = 0.f; efeat[p * 3 + 2] = 0.f;
  }
}

// ---------- MLP heads ----------
// eq1 = elu(concat(h[i], h[j], efeat) @ Wq1 + bq1); block = 64 threads per physical edge
__global__ void edge_mlp1(const float* __restrict__ H2, const float* __restrict__ efeat,
                          const int* __restrict__ pe, const float* __restrict__ Wq1,
                          const float* __restrict__ bq1, float* __restrict__ eq1) {
  __shared__ float ef[2 * DD1 + EDIM];   // 515
  int p = blockIdx.x;
  int i = pe[p * 2 + 0], j = pe[p * 2 + 1];
  for (int k = threadIdx.x; k < 2 * DD1 + EDIM; k += blockDim.x) {
    float v;
    if (k < DD1)            v = H2[(size_t)i * DD1 + k];
    else if (k < 2 * DD1)   v = H2[(size_t)j * DD1 + (k - DD1)];
    else                    v = efeat[p * 3 + (k - 2 * DD1)];
    ef[k] = v;
  }
  __syncthreads();
  int c = threadIdx.x;   // 0..63
  float acc = bq1[c];
  for (int k = 0; k < 2 * DD1 + EDIM; ++k) acc += ef[k] * Wq1[k * CC + c];
  eq1[(size_t)p * CC + c] = eluf(acc);
}

// out[r,c] = elu(in[r,:]@W + b);  blockDim = Cout, grid = rows
__global__ void mlp_elu(const float* __restrict__ in, const float* __restrict__ W,
                        const float* __restrict__ b, float* __restrict__ out,
                        int Kin, int Cout) {
  int r = blockIdx.x, c = threadIdx.x;
  const float* ip = in + (size_t)r * Kin;
  float acc = b[c];
  for (int k = 0; k < Kin; ++k) acc += ip[k] * W[k * Cout + c];
  out[(size_t)r * Cout + c] = eluf(acc);
}

// out[r] = in[r,:]@w + b[0]
__global__ void final_dot(const float* __restrict__ in, const float* __restrict__ w,
                          const float* __restrict__ b, float* __restrict__ out,
                          int rows, int K) {
  int r = blockIdx.x * blockDim.x + threadIdx.x;
  if (r >= rows) return;
  float acc = b[0];
  for (int k = 0; k < K; ++k) acc += in[(size_t)r * K + k] * w[k];
  out[r] = acc;
}

extern "C" void kernel_launch(void* const* d_in, const int* in_sizes, int n_in,
                              void* d_out, int out_size, void* d_ws, size_t ws_size,
                              hipStream_t stream) {
  const float* x     = (const float*)d_in[0];
  const int*   eidx  = (const int*)d_in[1];
  const float* eattr = (const float*)d_in[2];
  const int*   pe    = (const int*)d_in[3];
  const float *Wl1 = (const float*)d_in[4],  *bl1 = (const float*)d_in[5];
  const float *Wr1 = (const float*)d_in[6],  *br1 = (const float*)d_in[7];
  const float *We1 = (const float*)d_in[8],  *att1 = (const float*)d_in[9], *b1 = (const float*)d_in[10];
  const float *Wl2 = (const float*)d_in[11], *bl2 = (const float*)d_in[12];
  const float *Wr2 = (const float*)d_in[13], *br2 = (const float*)d_in[14];
  const float *We2 = (const float*)d_in[15], *att2 = (const float*)d_in[16], *b2 = (const float*)d_in[17];
  const float *Wq1 = (const float*)d_in[18], *bq1 = (const float*)d_in[19];
  const float *Wq2 = (const float*)d_in[20], *bq2 = (const float*)d_in[21];
  const float *Wq3 = (const float*)d_in[22], *bq3 = (const float*)d_in[23];
  const float *Wn1 = (const float*)d_in[24], *bn1 = (const float*)d_in[25];
  const float *Wn2 = (const float*)d_in[26], *bn2 = (const float*)d_in[27];
  const float *Wn3 = (const float*)d_in[28], *bn3 = (const float*)d_in[29];
  float* out = (float*)d_out;
  const int* src = eidx;
  const int* dst = eidx + EE;

  // Workspace carve-up (~50 MB)
  float* bufA   = (float*)d_ws;                 // NN*DD1
  float* bufB   = bufA + (size_t)NN * DD1;      // NN*DD1
  float* Hbuf   = bufB + (size_t)NN * DD1;      // NN*DD1
  float* H2     = Hbuf + (size_t)NN * DD1;      // NN*DD1
  float* logits = H2   + (size_t)NN * DD1;      // EE*HH
  float* den    = logits + (size_t)EE * HH;     // NN*HH
  unsigned* amaxU = (unsigned*)(den + (size_t)NN * HH);   // NN*HH
  unsigned* hkeys = amaxU + (size_t)NN * HH;    // HSIZE
  unsigned* hvals = hkeys + HSIZE;              // HSIZE
  float* efeat  = (float*)(hvals + HSIZE);      // PP*EDIM
  float* eq1    = efeat + (size_t)PP * EDIM;    // PP*CC
  float* eq2b   = eq1 + (size_t)PP * CC;        // PP*32

  const int tiles256 = (NN / 16) * (DD1 / 16);  // 10000 tiles
  const int grid256  = (tiles256 + 7) / 8;
  const int tiles64  = (NN / 16) * (CC / 16);   // 2500 tiles
  const int grid64   = (tiles64 + 7) / 8;

  // ===== GATv2 layer 1 =====
  wmma_gemm_f32<<<grid256, 256, 0, stream>>>(x, Wl1, bl1, bufA, NN, NDIM, DD1, 0);
  wmma_gemm_f32<<<grid256, 256, 0, stream>>>(x, Wr1, br1, bufB, NN, NDIM, DD1, 0);
  fill_u32<<<(NN * HH + 255) / 256, 256, 0, stream>>>(amaxU, ENC_NEG_INF, NN * HH);
  fill_f32<<<(NN * HH + 255) / 256, 256, 0, stream>>>(den, 0.f, NN * HH);
  bias_broadcast<<<NN, DD1, 0, stream>>>(Hbuf, b1);
  edge_logits<<<EE / 8, 256, 0, stream>>>(bufA, bufB, eattr, We1, att1, src, dst, logits, amaxU);
  edge_softmax_num<<<(EE * HH + 255) / 256, 256, 0, stream>>>(dst, logits, amaxU, den);
  edge_aggregate<<<EE, DD1, 0, stream>>>(src, dst, bufA, logits, den, Hbuf);
  elu_inplace<<<((NN * DD1) + 255) / 256, 256, 0, stream>>>(Hbuf, NN * DD1);

  // ===== GATv2 layer 2 =====
  wmma_gemm_f32<<<grid256, 256, 0, stream>>>(Hbuf, Wl2, bl2, bufA, NN, DD1, DD1, 0);
  wmma_gemm_f32<<<grid256, 256, 0, stream>>>(Hbuf, Wr2, br2, bufB, NN, DD1, DD1, 0);
  fill_u32<<<(NN * HH + 255) / 256, 256, 0, stream>>>(amaxU, ENC_NEG_INF, NN * HH);
  fill_f32<<<(NN * HH + 255) / 256, 256, 0, stream>>>(den, 0.f, NN * HH);
  bias_broadcast<<<NN, DD1, 0, stream>>>(H2, b2);
  edge_logits<<<EE / 8, 256, 0, stream>>>(bufA, bufB, eattr, We2, att2, src, dst, logits, amaxU);
  edge_softmax_num<<<(EE * HH + 255) / 256, 256, 0, stream>>>(dst, logits, amaxU, den);
  edge_aggregate<<<EE, DD1, 0, stream>>>(src, dst, bufA, logits, den, H2);
  elu_inplace<<<((NN * DD1) + 255) / 256, 256, 0, stream>>>(H2, NN * DD1);

  // ===== physical-edge attr lookup =====
  fill_u32<<<(HSIZE + 255) / 256, 256, 0, stream>>>(hkeys, HEMPTY, HSIZE);
  fill_u32<<<(HSIZE + 255) / 256, 256, 0, stream>>>(hvals, 0xFFFFFFFFu, HSIZE);
  hash_insert<<<(EE + 255) / 256, 256, 0, stream>>>(src, dst, hkeys, hvals);
  phys_lookup<<<(PP + 255) / 256, 256, 0, stream>>>(pe, hkeys, hvals, eattr, efeat);

  // ===== edge MLP head -> entangle_q (out[0:PP]) =====
  edge_mlp1<<<PP, CC, 0, stream>>>(H2, efeat, pe, Wq1, bq1, eq1);
  mlp_elu<<<PP, 32, 0, stream>>>(eq1, Wq2, bq2, eq2b, CC, 32);
  final_dot<<<(PP + 255) / 256, 256, 0, stream>>>(eq2b, Wq3, bq3, out, PP, 32);

  // ===== node MLP head -> swap_q (out[PP:PP+NN]) =====
  wmma_gemm_f32<<<grid64, 256, 0, stream>>>(H2, Wn1, bn1, bufA, NN, DD1, CC, 1);
  mlp_elu<<<NN, 32, 0, stream>>>(bufA, Wn2, bn2, bufB, CC, 32);
  final_dot<<<(NN + 255) / 256, 256, 0, stream>>>(bufB, Wn3, bn3, out + PP, NN, 32);
}